// LTConvLayer_29678224015511
// MI455X (gfx1250) — compile-verified
//
#include <hip/hip_runtime.h>
#include <stdint.h>

// ---------------------------------------------------------------------------
// LTConv LUT-tree kernel for gfx1250 (MI455X).
//   x  : [8, 64, 32, 32] f32
//   w0 : [32, 18, 16] f32   w1 : [32, 5, 16]   w2 : [32, 2, 16]   w3 : [32, 1, 16]
//   ci : [32, 8] i32
//   out: [8, 32, 32, 32] f32
// Block = (tree o, batch b, row-strip z of 8 rows), 256 threads = 256 pixels.
// TDM (tensor_load_to_lds) stages 8 channel row-strips + 4 weight tables into
// LDS; sigmoid applied once per staged element; LUT cells evaluated as
// 15-FMA lerp chains (mathematically identical to the corner-product einsum).
// ---------------------------------------------------------------------------

#define LT_B   8
#define LT_C   64
#define LT_H   32
#define LT_W   32
#define LT_O   32
#define LT_Q   8

#define TILE_ROWS   10           // up to 8 strip rows + 2 halo rows
#define TILE_STRIDE (TILE_ROWS * LT_W)   // 320 floats per channel tile
#define WBUF_W0     0            // 288 floats
#define WBUF_W1     288          // 80 floats
#define WBUF_W2     368          // 32 floats
#define WBUF_W3     400          // 16 floats
#define WBUF_TOT    416

#if __has_builtin(__builtin_amdgcn_tensor_load_to_lds)
#define LT_HAS_TDM 1
#else
#define LT_HAS_TDM 0
#endif

typedef uint32_t lt_u32x4 __attribute__((ext_vector_type(4)));
typedef int      lt_i32x4 __attribute__((ext_vector_type(4)));
typedef int      lt_i32x8 __attribute__((ext_vector_type(8)));

#if LT_HAS_TDM
// 1D contiguous f32 copy: global -> LDS via the Tensor Data Mover.
// D# group0: count=1 | lds_addr | global_addr[56:0] | type=2.
// D# group1: data_size=4B, tensor_dim0=tile_dim0=n, tensor_dim1=tile_dim1=1.
__device__ __forceinline__ void lt_tdm_copy_f32(const void* gptr,
                                                uint32_t lds_byte_addr,
                                                uint32_t n) {
  uint64_t ga = (uint64_t)(uintptr_t)gptr;
  lt_u32x4 g0;
  g0[0] = 1u;                                             // count=1, gather off
  g0[1] = lds_byte_addr;                                  // LDS byte address
  g0[2] = (uint32_t)(ga & 0xFFFFFFFFu);                   // global_addr[31:0]
  g0[3] = (uint32_t)((ga >> 32) & 0x01FFFFFFu) | (2u << 30); // addr[56:32]|type=2
  lt_i32x8 g1;
  g1[0] = 0x00020000;                                     // data_size=2 (4 bytes)
  g1[1] = (int)((n & 0xFFFFu) << 16);                     // tensor_dim0[15:0]
  g1[2] = (int)(((n >> 16) & 0xFFFFu) | (1u << 16));      // dim0 hi | tensor_dim1=1
  g1[3] = (int)((n & 0xFFFFu) << 16);                     // tile_dim0 = n
  g1[4] = 1;                                              // tile_dim1 = 1
  g1[5] = (int)n;                                         // tensor_dim0_stride
  g1[6] = 0;
  g1[7] = 0;
  lt_i32x4 z4 = {0, 0, 0, 0};
#if defined(__clang_major__) && (__clang_major__ >= 23)
  lt_i32x8 z8 = {0, 0, 0, 0, 0, 0, 0, 0};
  __builtin_amdgcn_tensor_load_to_lds(g0, g1, z4, z4, z8, 0);
#else
  __builtin_amdgcn_tensor_load_to_lds(g0, g1, z4, z4, 0);
#endif
}
#endif

__device__ __forceinline__ float lt_sigmoid(float v) {
  return 1.0f / (1.0f + __expf(-v));
}

// 4-input multilinear LUT cell: contract 16 corner weights with the
// tensor-product basis of (1-x_i, x_i) via a lerp chain (bit i <-> input i).
__device__ __forceinline__ float lt_lut4(const float* __restrict__ wt,
                                         const float* __restrict__ xin) {
  float t[16];
#pragma unroll
  for (int c = 0; c < 16; ++c) t[c] = wt[c];
#pragma unroll
  for (int j = 0; j < 8; ++j) t[j] = __builtin_fmaf(xin[3], t[j + 8] - t[j], t[j]);
#pragma unroll
  for (int j = 0; j < 4; ++j) t[j] = __builtin_fmaf(xin[2], t[j + 4] - t[j], t[j]);
#pragma unroll
  for (int j = 0; j < 2; ++j) t[j] = __builtin_fmaf(xin[1], t[j + 2] - t[j], t[j]);
  return __builtin_fmaf(xin[0], t[1] - t[0], t[0]);
}

__global__ void __launch_bounds__(256)
LTConvLayer_kernel(const float* __restrict__ x,
                   const float* __restrict__ w0,
                   const float* __restrict__ w1,
                   const float* __restrict__ w2,
                   const float* __restrict__ w3,
                   const int* __restrict__ ci,
                   float* __restrict__ out) {
  __shared__ float tiles[LT_Q * TILE_STRIDE];   // 8 x 320 f32 = 10 KB
  __shared__ float wbuf[WBUF_TOT];              // 416 f32

  const int o   = blockIdx.x;       // tree
  const int b   = blockIdx.y;       // batch
  const int h0  = blockIdx.z * 8;   // first output row of this strip
  const int tid = threadIdx.x;

  // Row-strip staged from global: rows [r0, rend) of the 32x32 plane.
  const int r0   = (h0 - 1 < 0) ? 0 : (h0 - 1);
  const int rend = (h0 + 9 > LT_H) ? LT_H : (h0 + 9);
  const int rn   = rend - r0;               // 9 or 10 rows
  const int nstage = rn * LT_W;             // floats per channel tile

#if LT_HAS_TDM
  if (tid < 32) {                           // wave 0 drives the TDM
    uint32_t tbase = (uint32_t)(uintptr_t)(&tiles[0]);
    uint32_t wbase = (uint32_t)(uintptr_t)(&wbuf[0]);
#pragma unroll
    for (int q = 0; q < LT_Q; ++q) {
      int c = __builtin_amdgcn_readfirstlane(ci[o * LT_Q + q]);
      const float* src = x + (((size_t)b * LT_C + c) * (LT_H * LT_W) + r0 * LT_W);
      lt_tdm_copy_f32(src, tbase + (uint32_t)(q * TILE_STRIDE) * 4u, (uint32_t)nstage);
    }
    lt_tdm_copy_f32(w0 + (size_t)o * 288, wbase + WBUF_W0 * 4u, 288);
    lt_tdm_copy_f32(w1 + (size_t)o * 80,  wbase + WBUF_W1 * 4u, 80);
    lt_tdm_copy_f32(w2 + (size_t)o * 32,  wbase + WBUF_W2 * 4u, 32);
    lt_tdm_copy_f32(w3 + (size_t)o * 16,  wbase + WBUF_W3 * 4u, 16);
#if __has_builtin(__builtin_amdgcn_s_wait_tensorcnt)
    __builtin_amdgcn_s_wait_tensorcnt(0);
#else
    asm volatile("s_wait_tensorcnt 0x0" ::: "memory");
#endif
  }
  __syncthreads();
#else
  // Fallback: cooperative vector loads.
  for (int q = 0; q < LT_Q; ++q) {
    int c = ci[o * LT_Q + q];
    const float* src = x + (((size_t)b * LT_C + c) * (LT_H * LT_W) + r0 * LT_W);
    for (int i = tid; i < nstage; i += 256) tiles[q * TILE_STRIDE + i] = src[i];
  }
  for (int i = tid; i < 288; i += 256) wbuf[WBUF_W0 + i] = w0[(size_t)o * 288 + i];
  for (int i = tid; i < 80;  i += 256) wbuf[WBUF_W1 + i] = w1[(size_t)o * 80 + i];
  for (int i = tid; i < 32;  i += 256) wbuf[WBUF_W2 + i] = w2[(size_t)o * 32 + i];
  for (int i = tid; i < 16;  i += 256) wbuf[WBUF_W3 + i] = w3[(size_t)o * 16 + i];
  __syncthreads();
#endif

  // Sigmoid once per staged element (deduplicates the 9x tap reuse).
  const int total = LT_Q * TILE_STRIDE;
  for (int i = tid; i < total; i += 256) {
    // Rows beyond rn are never read; converting them is harmless.
    tiles[i] = lt_sigmoid(tiles[i]);
  }
  __syncthreads();

  // One thread = one output pixel of this strip.
  const int h = h0 + (tid >> 5);
  const int w = tid & 31;

  // ---- Level 0: 18 cells over 72 inputs (q, ky, kx), OOB taps -> 0.5 ----
  float lvl0[18];
#pragma unroll
  for (int l = 0; l < 18; ++l) {
    float xin[4];
#pragma unroll
    for (int ii = 0; ii < 4; ++ii) {
      const int j  = 4 * l + ii;        // input index in [0,72)
      const int q  = j / 9;
      const int r  = j - q * 9;
      const int ky = r / 3;
      const int kx = r - ky * 3;
      const int hy = h + ky - 1;
      const int wx = w + kx - 1;
      const bool inb = (hy >= 0) & (hy < LT_H) & (wx >= 0) & (wx < LT_W);
      xin[ii] = inb ? tiles[q * TILE_STRIDE + (hy - r0) * LT_W + wx] : 0.5f;
    }
    lvl0[l] = lt_lut4(&wbuf[WBUF_W0 + l * 16], xin);
  }

  // ---- Level 1: sigmoid, pad 18 -> 20 with 0.5, 5 cells ----
  float in1[20];
#pragma unroll
  for (int j = 0; j < 18; ++j) in1[j] = lt_sigmoid(lvl0[j]);
  in1[18] = 0.5f;
  in1[19] = 0.5f;
  float lvl1[5];
#pragma unroll
  for (int l = 0; l < 5; ++l)
    lvl1[l] = lt_lut4(&wbuf[WBUF_W1 + l * 16], &in1[4 * l]);

  // ---- Level 2: sigmoid, pad 5 -> 8 with 0.5, 2 cells ----
  float in2[8];
#pragma unroll
  for (int j = 0; j < 5; ++j) in2[j] = lt_sigmoid(lvl1[j]);
  in2[5] = in2[6] = in2[7] = 0.5f;
  float lvl2[2];
#pragma unroll
  for (int l = 0; l < 2; ++l)
    lvl2[l] = lt_lut4(&wbuf[WBUF_W2 + l * 16], &in2[4 * l]);

  // ---- Level 3: sigmoid, pad 2 -> 4 with 0.5, 1 cell, no final sigmoid ----
  float in3[4];
  in3[0] = lt_sigmoid(lvl2[0]);
  in3[1] = lt_sigmoid(lvl2[1]);
  in3[2] = 0.5f;
  in3[3] = 0.5f;
  const float y = lt_lut4(&wbuf[WBUF_W3], in3);

  // out: [B, O, H, W]
  out[((((size_t)b * LT_O + o) * LT_H) + h) * LT_W + w] = y;
}

extern "C" void kernel_launch(void* const* d_in, const int* in_sizes, int n_in,
                              void* d_out, int out_size, void* d_ws, size_t ws_size,
                              hipStream_t stream) {
  (void)in_sizes; (void)n_in; (void)out_size; (void)d_ws; (void)ws_size;
  const float* x  = (const float*)d_in[0];
  const float* w0 = (const float*)d_in[1];
  const float* w1 = (const float*)d_in[2];
  const float* w2 = (const float*)d_in[3];
  const float* w3 = (const float*)d_in[4];
  const int*   ci = (const int*)d_in[5];
  float* out = (float*)d_out;

  dim3 grid(LT_O, LT_B, LT_H / 8);   // (32, 8, 4) = 1024 blocks
  dim3 block(256);                   // 8 wave32 waves
  LTConvLayer_kernel<<<grid, block, 0, stream>>>(x, w0, w1, w2, w3, ci, out);
}